// KnapsackGNN_1477468750494
// MI455X (gfx1250) — compile-verified
//
#include <hip/hip_runtime.h>

// GCN forward for MI455X (gfx1250, wave32). Dense transforms use
// V_WMMA_F32_16X16X4_F32 (fp32 matrix pipe, K=4 == IN_DIM); sparse
// aggregation uses non-returning global_atomic_add_f32 (L2 atomics;
// scatter is HBM/L2 bound at ~23.3 TB/s; working set fits in 192MB L2).
//
// Codegen notes vs previous round: uniform full-tile fast paths remove the
// per-row exec-mask/store branching; W2 is LDS-packed as float2 K-pairs so
// each WMMA K-step needs exactly one ds_load_b64 + one global_load_b64.

typedef float v2f __attribute__((ext_vector_type(2)));
typedef float v8f __attribute__((ext_vector_type(8)));

#define IN_DIM 4
#define HID 64

// ---------------- degree / normalization ----------------

__global__ void deg_init_kernel(float* __restrict__ deg, int n) {
  int i = blockIdx.x * blockDim.x + threadIdx.x;
  if (i < n) deg[i] = 1.0f;  // self loop contributes 1
}

__global__ void deg_scatter_kernel(const int* __restrict__ dst,
                                   float* __restrict__ deg, int ne) {
  int e = blockIdx.x * blockDim.x + threadIdx.x;
  if (e < ne) unsafeAtomicAdd(&deg[dst[e]], 1.0f);
}

__global__ void rsqrt_kernel(float* __restrict__ dinv, int n) {
  int i = blockIdx.x * blockDim.x + threadIdx.x;
  if (i < n) dinv[i] = __frsqrt_rn(dinv[i]);
}

// ---------------- dense transforms via WMMA ----------------

// h = x @ W1 ; x:[n,4], W1:[4,64]. One wave = 16 nodes x 16 cols, single
// v_wmma_f32_16x16x4_f32 (K = IN_DIM = 4). Block = 128 threads = 4 waves
// covering the 64 output columns of a 16-node tile.
__global__ void gemm1_wmma_kernel(const float* __restrict__ x,
                                  const float* __restrict__ W1,
                                  float* __restrict__ h, int n) {
  const int lane = threadIdx.x & 31;
  const int wv   = threadIdx.x >> 5;      // 0..3 : 16-col tile
  const int m    = lane & 15;
  const int hi   = lane >> 4;             // 0 or 1 (K half / M half)
  const int row0 = blockIdx.x * 16;
  const int node = row0 + m;
  const int col  = wv * 16 + m;

  // B 4x16 fp32: v0 = B[2*hi][col], v1 = B[2*hi+1][col]
  v2f b;
  b.x = W1[(2 * hi + 0) * HID + col];
  b.y = W1[(2 * hi + 1) * HID + col];

  if (row0 + 16 <= n) {                   // uniform fast path (full tile)
    const float2 av = *(const float2*)(x + node * IN_DIM + 2 * hi);
    v2f a; a.x = av.x; a.y = av.y;
    v8f c = {};
    c = __builtin_amdgcn_wmma_f32_16x16x4_f32(false, a, false, b,
                                              (short)0, c, false, false);
#pragma unroll
    for (int r = 0; r < 8; ++r)
      h[(row0 + r + 8 * hi) * HID + col] = c[r];
  } else {                                // tail tile (guarded)
    v2f a;
    if (node < n) {
      a.x = x[node * IN_DIM + 2 * hi + 0];
      a.y = x[node * IN_DIM + 2 * hi + 1];
    } else { a.x = 0.0f; a.y = 0.0f; }
    v8f c = {};
    c = __builtin_amdgcn_wmma_f32_16x16x4_f32(false, a, false, b,
                                              (short)0, c, false, false);
#pragma unroll
    for (int r = 0; r < 8; ++r) {
      int row = row0 + r + 8 * hi;
      if (row < n) h[row * HID + col] = c[r];
    }
  }
}

// h_out = h_in @ W2 ; [n,64] @ [64,64]. W2 staged in LDS as float2 K-pairs
// (16KB). Block = 256 threads = 8 waves = 2 node-tiles x 4 col-tiles; each
// wave runs a 16-iteration K-loop of v_wmma_f32_16x16x4_f32 with one
// global_load_b64 (A pair) + one ds_load_b64 (B pair) per step.
__global__ void gemm2_wmma_kernel(const float* __restrict__ h_in,
                                  const float* __restrict__ W2,
                                  float* __restrict__ h_out, int n) {
  // sP[p][col] = { W2[2p][col], W2[2p+1][col] }, p = K-pair index 0..31
  __shared__ float2 sP[(HID / 2) * HID];
  for (int i = threadIdx.x; i < (HID / 2) * HID; i += blockDim.x) {
    int p = i >> 6;
    int cc = i & 63;
    sP[i] = make_float2(W2[(2 * p + 0) * HID + cc],
                        W2[(2 * p + 1) * HID + cc]);
  }
  __syncthreads();

  const int lane = threadIdx.x & 31;
  const int wid  = threadIdx.x >> 5;   // 0..7
  const int mt   = wid >> 2;           // node tile 0..1
  const int nt   = wid & 3;            // col tile 0..3
  const int m    = lane & 15;
  const int hi   = lane >> 4;
  const int row0 = blockIdx.x * 32 + mt * 16;
  const int node = row0 + m;
  const int col  = nt * 16 + m;

  v8f c = {};
  if (row0 + 16 <= n) {                 // uniform fast path (full tile)
    const float2* arow = (const float2*)(h_in + node * HID);
#pragma unroll
    for (int k0 = 0; k0 < HID; k0 += 4) {
      const int p = (k0 >> 1) + hi;     // K-pair index: ka = 2p
      const float2 av = arow[p];
      const float2 bv = sP[(p << 6) + col];
      v2f a; a.x = av.x; a.y = av.y;
      v2f b; b.x = bv.x; b.y = bv.y;
      c = __builtin_amdgcn_wmma_f32_16x16x4_f32(false, a, false, b,
                                                (short)0, c, false, false);
    }
#pragma unroll
    for (int r = 0; r < 8; ++r)
      h_out[(row0 + r + 8 * hi) * HID + col] = c[r];
  } else {                              // tail tile (guarded)
#pragma unroll
    for (int k0 = 0; k0 < HID; k0 += 4) {
      const int p = (k0 >> 1) + hi;
      v2f a;
      if (node < n) {
        a.x = h_in[node * HID + 2 * p + 0];
        a.y = h_in[node * HID + 2 * p + 1];
      } else { a.x = 0.0f; a.y = 0.0f; }
      const float2 bv = sP[(p << 6) + col];
      v2f b; b.x = bv.x; b.y = bv.y;
      c = __builtin_amdgcn_wmma_f32_16x16x4_f32(false, a, false, b,
                                                (short)0, c, false, false);
    }
#pragma unroll
    for (int r = 0; r < 8; ++r) {
      int row = row0 + r + 8 * hi;
      if (row < n) h_out[row * HID + col] = c[r];
    }
  }
}

// ---------------- sparse aggregation ----------------

// agg[i][c] = h[i][c] * dinv[i]^2   (self-loop term of (A+I) normalization)
__global__ void agg_init_kernel(const float* __restrict__ h,
                                const float* __restrict__ dinv,
                                float* __restrict__ agg, int n) {
  int i = blockIdx.x * blockDim.x + threadIdx.x;
  if (i < n * HID) {
    float di = dinv[i >> 6];  // HID == 64
    agg[i] = h[i] * di * di;
  }
}

// agg[dst] += h[src] * dinv[src]*dinv[dst] ; one wave per edge, each lane
// handles channels {lane, lane+32} -> 64 f32 L2 atomics per edge.
__global__ void edge_scatter_kernel(const int* __restrict__ src,
                                    const int* __restrict__ dst,
                                    const float* __restrict__ dinv,
                                    const float* __restrict__ h,
                                    float* __restrict__ agg, int ne) {
  int e = blockIdx.x * (blockDim.x >> 5) + (threadIdx.x >> 5);
  if (e >= ne) return;
  int lane = threadIdx.x & 31;
  int s = src[e];
  int d = dst[e];
  float norm = dinv[s] * dinv[d];
  float v0 = h[s * HID + lane]      * norm;
  float v1 = h[s * HID + lane + 32] * norm;
  unsafeAtomicAdd(&agg[d * HID + lane],      v0);
  unsafeAtomicAdd(&agg[d * HID + lane + 32], v1);
}

__global__ void bias_relu_kernel(float* __restrict__ agg,
                                 const float* __restrict__ bias, int n) {
  int i = blockIdx.x * blockDim.x + threadIdx.x;
  if (i < n * HID) agg[i] = fmaxf(agg[i] + bias[i & (HID - 1)], 0.0f);
}

// ---------------- readout: logits = h @ Wl + bl ----------------

__global__ void readout_kernel(const float* __restrict__ h,
                               const float* __restrict__ Wl,
                               const float* __restrict__ bl,
                               float* __restrict__ out, int n) {
  int node = blockIdx.x * (blockDim.x >> 5) + (threadIdx.x >> 5);
  if (node >= n) return;
  int lane = threadIdx.x & 31;
  float s = h[node * HID + lane]      * Wl[lane] +
            h[node * HID + lane + 32] * Wl[lane + 32];
#pragma unroll
  for (int off = 16; off > 0; off >>= 1) s += __shfl_down(s, off, 32);
  if (lane == 0) out[node] = s + bl[0];
}

// ---------------- driver ----------------

extern "C" void kernel_launch(void* const* d_in, const int* in_sizes, int n_in,
                              void* d_out, int out_size, void* d_ws, size_t ws_size,
                              hipStream_t stream) {
  (void)n_in; (void)out_size; (void)ws_size;

  const float* x  = (const float*)d_in[0];
  const int*   ei = (const int*)d_in[1];   // edge_index [2, E]
  const float* W1 = (const float*)d_in[2];
  const float* b1 = (const float*)d_in[3];
  const float* W2 = (const float*)d_in[4];
  const float* b2 = (const float*)d_in[5];
  const float* Wl = (const float*)d_in[6];
  const float* bl = (const float*)d_in[7];
  float* out = (float*)d_out;

  const int n  = in_sizes[0] / IN_DIM;
  const int ne = in_sizes[1] / 2;
  const int* src = ei;
  const int* dst = ei + ne;

  // workspace layout: dinv[n] | bufA[n*64] | bufB[n*64]
  char* ws = (char*)d_ws;
  size_t off = 0;
  float* dinv = (float*)(ws + off);
  off += (((size_t)n * 4) + 255) & ~(size_t)255;
  float* bufA = (float*)(ws + off);                 // h1, then h2
  off += (((size_t)n * HID * 4) + 255) & ~(size_t)255;
  float* bufB = (float*)(ws + off);                 // agg1/h1_act, agg2/h2_act

  const int T = 256;
  const int gN   = (n + T - 1) / T;
  const int gE   = (ne + T - 1) / T;
  const int gNH  = (n * HID + T - 1) / T;
  const int gE8  = (ne + 7) / 8;   // 8 waves (edges) per 256-thread block
  const int gN8  = (n + 7) / 8;    // 8 waves (nodes) per 256-thread block
  const int gM16 = (n + 15) / 16;
  const int gM32 = (n + 31) / 32;

  // normalization: deg -> dinv
  deg_init_kernel<<<gN, T, 0, stream>>>(dinv, n);
  deg_scatter_kernel<<<gE, T, 0, stream>>>(dst, dinv, ne);
  rsqrt_kernel<<<gN, T, 0, stream>>>(dinv, n);

  // layer 1
  gemm1_wmma_kernel<<<gM16, 128, 0, stream>>>(x, W1, bufA, n);
  agg_init_kernel<<<gNH, T, 0, stream>>>(bufA, dinv, bufB, n);
  edge_scatter_kernel<<<gE8, T, 0, stream>>>(src, dst, dinv, bufA, bufB, ne);
  bias_relu_kernel<<<gNH, T, 0, stream>>>(bufB, b1, n);

  // layer 2 (bufB = h1_act consumed by GEMM before being overwritten)
  gemm2_wmma_kernel<<<gM32, T, 0, stream>>>(bufB, W2, bufA, n);
  agg_init_kernel<<<gNH, T, 0, stream>>>(bufA, dinv, bufB, n);
  edge_scatter_kernel<<<gE8, T, 0, stream>>>(src, dst, dinv, bufA, bufB, ne);
  bias_relu_kernel<<<gNH, T, 0, stream>>>(bufB, b2, n);

  // readout
  readout_kernel<<<gN8, T, 0, stream>>>(bufB, Wl, bl, out, n);
}